// GetPoseDetectionBBNN_45870250721463
// MI455X (gfx1250) — compile-verified
//
#include <hip/hip_runtime.h>
#include <math.h>

#define BLOCK 128
#define NTH   2100            // 8400 anchors / 4 pixels per thread

__device__ __forceinline__ float sigf(float x) { return 1.0f / (1.0f + expf(-x)); }

#if __has_builtin(__builtin_amdgcn_s_wait_asynccnt)
#define WAIT_ASYNC(n) __builtin_amdgcn_s_wait_asynccnt(n)
#else
#define WAIT_ASYNC(n) asm volatile("s_wait_asynccnt %0" :: "i"(n) : "memory")
#endif
#if __has_builtin(__builtin_amdgcn_s_wait_dscnt)
#define WAIT_DS(n) __builtin_amdgcn_s_wait_dscnt(n)
#else
#define WAIT_DS(n) asm volatile("s_wait_dscnt %0" :: "i"(n) : "memory")
#endif

__device__ __forceinline__ void async_ld_b128(unsigned int lds_addr, const void* gptr) {
    unsigned long long ga = (unsigned long long)(size_t)gptr;
    asm volatile("global_load_async_to_lds_b128 %0, %1, off"
                 :: "v"(lds_addr), "v"(ga) : "memory");
}

// Re-init the reduction cell every call with sentinel-0's key (score 0.47, idx 8400 -> enc 1).
__global__ void k_init(unsigned long long* best) {
    best[0] = ((unsigned long long)__float_as_uint(0.47f) << 32) | 1ull;
}

__global__ __launch_bounds__(BLOCK) void k_reduce(
    const float* __restrict__ cls0, const float* __restrict__ cls1, const float* __restrict__ cls2,
    const float* __restrict__ obj0, const float* __restrict__ obj1, const float* __restrict__ obj2,
    unsigned long long* __restrict__ best)
{
    __shared__ float4 buf[2][BLOCK];     // 4 KB double buffer, one float4 slot per thread
    const int t = blockIdx.x * BLOCK + threadIdx.x;
    if (t >= NTH) return;

    const float* cls; const float* obj; int HW, p0, gbase;
    if (t < 1600)      { cls = cls0; obj = obj0; HW = 6400; p0 = t << 2;          gbase = 0;    }
    else if (t < 2000) { cls = cls1; obj = obj1; HW = 1600; p0 = (t - 1600) << 2; gbase = 6400; }
    else               { cls = cls2; obj = obj2; HW = 400;  p0 = (t - 2000) << 2; gbase = 8000; }

    const unsigned int lds0 = (unsigned int)(size_t)&buf[0][threadIdx.x];
    const unsigned int lds1 = (unsigned int)(size_t)&buf[1][threadIdx.x];

    // Prologue: stage channels 0 and 1 (2 async ops in flight).
    async_ld_b128(lds0, cls + p0);
    async_ld_b128(lds1, cls + HW + p0);

    float4 mx, l0;
    #pragma unroll 1
    for (int c = 0; c < 80; ++c) {
        if (c < 78) WAIT_ASYNC(1);   // channel c landed (c+1 still in flight)
        else        WAIT_ASYNC(0);   // tail: drain

        float4 v = buf[c & 1][threadIdx.x];          // ds_load_b128

        if (c + 2 < 80) {
            WAIT_DS(0);                              // WAR: ds read done before DMA rewrites slot
            async_ld_b128((c & 1) ? lds1 : lds0,
                          cls + (size_t)(c + 2) * HW + p0);
        }

        if (c == 0) { l0 = v; mx = v; }
        else {
            mx.x = fmaxf(mx.x, v.x); mx.y = fmaxf(mx.y, v.y);
            mx.z = fmaxf(mx.z, v.z); mx.w = fmaxf(mx.w, v.w);
        }
    }

    const float4 o4 = *reinterpret_cast<const float4*>(obj + p0);
    const float ox[4] = { o4.x, o4.y, o4.z, o4.w };
    const float mv[4] = { mx.x, mx.y, mx.z, mx.w };
    const float lv[4] = { l0.x, l0.y, l0.z, l0.w };

    unsigned long long bestk = 0;
    #pragma unroll
    for (int j = 0; j < 4; ++j) {
        // argmax class == 0  <=>  channel-0 logit equals the row max
        if (lv[j] >= mv[j]) {
            float sc = sigf(ox[j]) * sigf(mv[j]);    // == max_sc for this row
            if (sc > 0.1f) {
                unsigned long long k =
                    ((unsigned long long)__float_as_uint(sc) << 32) |
                    (unsigned)(8401 - (gbase + p0 + j));   // larger enc = smaller idx
                bestk = bestk > k ? bestk : k;
            }
        }
    }
    if (bestk) atomicMax(best, bestk);
}

__global__ void k_final(const float* __restrict__ box0, const float* __restrict__ box1,
                        const float* __restrict__ box2,
                        const unsigned long long* __restrict__ best, float* __restrict__ out)
{
    const unsigned long long k = best[0];
    const float sc = __uint_as_float((unsigned)(k >> 32));
    const int idx = 8401 - (int)(unsigned)(k & 0xffffffffu);
    float b0, b1, b2, b3;
    if (idx >= 8400) {                                // sentinel rows
        if (idx == 8400) { b0 = 0.f;   b1 = 0.f;   b2 = 640.f; b3 = 640.f; }
        else             { b0 = 320.f; b1 = 320.f; b2 = 540.f; b3 = 540.f; }
    } else {
        const float* box; int W, HW, pl; float st;
        if (idx < 6400)      { box = box0; W = 80; HW = 6400; pl = idx;        st = 8.f;  }
        else if (idx < 8000) { box = box1; W = 40; HW = 1600; pl = idx - 6400; st = 16.f; }
        else                 { box = box2; W = 20; HW = 400;  pl = idx - 8000; st = 32.f; }
        const float bx = box[pl], by = box[HW + pl], bw = box[2 * HW + pl], bh = box[3 * HW + pl];
        const float gx = (float)(pl % W), gy = (float)(pl / W);
        const float cx = (bx + gx) * st, cy = (by + gy) * st;
        const float w = expf(bw) * st,   h = expf(bh) * st;
        b0 = cx - 0.5f * w; b1 = cy - 0.5f * h; b2 = cx + 0.5f * w; b3 = cy + 0.5f * h;
    }
    out[0] = b0; out[1] = b1; out[2] = b2; out[3] = b3; out[4] = sc;
}

extern "C" void kernel_launch(void* const* d_in, const int* in_sizes, int n_in,
                              void* d_out, int out_size, void* d_ws, size_t ws_size,
                              hipStream_t stream)
{
    const float* cls0 = (const float*)d_in[0];
    const float* cls1 = (const float*)d_in[1];
    const float* cls2 = (const float*)d_in[2];
    const float* box0 = (const float*)d_in[3];
    const float* box1 = (const float*)d_in[4];
    const float* box2 = (const float*)d_in[5];
    const float* obj0 = (const float*)d_in[6];
    const float* obj1 = (const float*)d_in[7];
    const float* obj2 = (const float*)d_in[8];

    unsigned long long* best = (unsigned long long*)d_ws;
    float* out = (float*)d_out;

    k_init<<<1, 1, 0, stream>>>(best);
    k_reduce<<<(NTH + BLOCK - 1) / BLOCK, BLOCK, 0, stream>>>(
        cls0, cls1, cls2, obj0, obj1, obj2, best);
    k_final<<<1, 1, 0, stream>>>(box0, box1, box2, best, out);
}